// NumAwareFeatureNetwork_37031208026259
// MI455X (gfx1250) — compile-verified
//
#include <hip/hip_runtime.h>

typedef float v2f __attribute__((ext_vector_type(2)));
typedef float v8f __attribute__((ext_vector_type(8)));

#define VOCAB   32000
#define HIDDEN  1024
#define BATCH   256
#define SEQ     2048

// One wave32 owns a 16-wide hidden tile of one batch row and walks the whole
// sequence.  V_WMMA_F32_16X16X4_F32 accumulates D = A(mask-broadcast) x
// B(gathered emb rows) + C, 4 sequence steps per instruction, fully in f32.
// A 16x4 layout: VGPR0 = {K0 lanes0-15 | K2 lanes16-31}, VGPR1 = {K1 | K3}.
// B 4x16 layout: VGPR0 = {row s0 lanes0-15 | row s2 lanes16-31}, VGPR1 = {s1 | s3}.
// Every M-row of D is identical => each lane's acc[0] is the pooled feature
// for hidden column (lane & 15); no cross-lane reduction needed.
__global__ __launch_bounds__(256) void numaware_pool_wmma_kernel(
    const int*   __restrict__ ids,        // (B, S) int32
    const float* __restrict__ numvals,    // (B, S)
    const float* __restrict__ mask,       // (B, S)
    const float* __restrict__ emb,        // (VOCAB, HIDDEN)
    const float* __restrict__ w_num,      // (HIDDEN)
    const float* __restrict__ b_num,      // (HIDDEN)
    const int*   __restrict__ ntid_p,     // scalar
    float*       __restrict__ out)        // (B, HIDDEN)
{
    const int b    = blockIdx.y;
    const int lane = (int)(threadIdx.x & 31u);
    const int wave = (int)(threadIdx.x >> 5u);
    const int h    = blockIdx.x * 128 + wave * 16 + (lane & 15);
    const bool lo  = lane < 16;

    const int*   idrow = ids  + (size_t)b * SEQ;
    const float* mrow  = mask + (size_t)b * SEQ;

    v8f acc0 = {};
    v8f acc1 = {};
    float msum = 0.0f;

    for (int s = 0; s < SEQ; s += 8) {
        // Workgroup-uniform scalar loads of ids + mask for 8 sequence steps.
        int4   i4a = *(const int4*)  (idrow + s);
        int4   i4b = *(const int4*)  (idrow + s + 4);
        float4 m4a = *(const float4*)(mrow  + s);
        float4 m4b = *(const float4*)(mrow  + s + 4);

        // Stream the uniform id/mask arrays ~1KB ahead (global_prefetch_b8).
        __builtin_prefetch(idrow + s + 256, 0, 1);
        __builtin_prefetch(mrow  + s + 256, 0, 1);

        {   // sequence steps s .. s+3
            int ia = lo ? i4a.x : i4a.z;
            int ib = lo ? i4a.y : i4a.w;
            v2f A;  A.x  = lo ? m4a.x : m4a.z;  A.y  = lo ? m4a.y : m4a.w;
            v2f Bv; Bv.x = emb[((long)ia << 10) + h];
                    Bv.y = emb[((long)ib << 10) + h];
            acc0 = __builtin_amdgcn_wmma_f32_16x16x4_f32(
                false, A, false, Bv, (short)0, acc0, false, false);
        }
        {   // sequence steps s+4 .. s+7 (independent accumulator chain)
            int ia = lo ? i4b.x : i4b.z;
            int ib = lo ? i4b.y : i4b.w;
            v2f A;  A.x  = lo ? m4b.x : m4b.z;  A.y  = lo ? m4b.y : m4b.w;
            v2f Bv; Bv.x = emb[((long)ia << 10) + h];
                    Bv.y = emb[((long)ib << 10) + h];
            acc1 = __builtin_amdgcn_wmma_f32_16x16x4_f32(
                false, A, false, Bv, (short)0, acc1, false, false);
        }

        msum += (m4a.x + m4a.y) + (m4a.z + m4a.w)
              + (m4b.x + m4b.y) + (m4b.z + m4b.w);
    }

    v8f acc = acc0 + acc1;                  // elementwise: layouts identical
    float denom = fmaxf(msum, 1.0f);
    float feat  = acc[0] * (1.0f / denom);  // all 8 acc regs hold the same value

    if (lo) {                               // lanes 16-31 are duplicates
        float w     = w_num[h];
        float bb    = b_num[h];
        float lastv = numvals[(size_t)b * SEQ + (SEQ - 1)];
        float z     = fmaf(lastv, w, bb);
        float g     = 1.0f / (1.0f + __expf(-z));
        bool isnum  = (idrow[SEQ - 1] == ntid_p[0]);
        out[(size_t)b * HIDDEN + h] = isnum ? feat * g : feat;
    }
}

extern "C" void kernel_launch(void* const* d_in, const int* in_sizes, int n_in,
                              void* d_out, int out_size, void* d_ws, size_t ws_size,
                              hipStream_t stream) {
    const int*   ids     = (const int*)  d_in[0];
    const float* numvals = (const float*)d_in[1];
    const float* amask   = (const float*)d_in[2];
    const float* emb     = (const float*)d_in[3];
    const float* w_num   = (const float*)d_in[4];
    const float* b_num   = (const float*)d_in[5];
    const int*   ntid    = (const int*)  d_in[6];
    float*       out     = (float*)d_out;

    dim3 grid(HIDDEN / 128, BATCH);   // (8, 256): 8 hidden tiles x 256 batch rows
    dim3 block(256);                  // 8 wave32s, each owns a 16-wide hidden tile
    numaware_pool_wmma_kernel<<<grid, block, 0, stream>>>(
        ids, numvals, amask, emb, w_num, b_num, ntid, out);
}